// LSTMCaptioning_Attention_8418135900477
// MI455X (gfx1250) — compile-verified
//
#include <hip/hip_runtime.h>
#include <cmath>

// Model dims
#define BSZ   128
#define PIX   196
#define EDIM  512
#define HDIM  512
#define ADIM  512
#define EMBD  512
#define VOC   32000
#define TSTEP 20
#define NGATE 2048      // 4*H
#define NPART 500       // VOC/64 logits column blocks

typedef __attribute__((ext_vector_type(2))) float v2f;
typedef __attribute__((ext_vector_type(8))) float v8f;

__device__ __forceinline__ float sigmoidf_(float x) { return 1.0f / (1.0f + __expf(-x)); }

// ---------------------------------------------------------------------------
// WMMA helper: accumulate a K=512 contraction into 4 stacked 16x16 tiles.
//   C[m0 + i*16 .. +15][n0 .. n0+15] += A[m, k] * W[n, k]   (B = W^T)
// A fragment layout (16x4 f32): lanes 0-15 hold M=lane, K={0,1}; lanes 16-31
// hold M=lane-16, K={2,3}. B fragment mirrors with N=lane&15. Both operands
// therefore load as contiguous float2 from row-major (.,512) matrices.
// ---------------------------------------------------------------------------
__device__ __forceinline__ void wmma_k512(v8f acc[4],
                                          const float* __restrict__ Abase,
                                          const float* __restrict__ Wrow,
                                          int m0, int mh, int kk) {
  #pragma unroll 4
  for (int k0 = 0; k0 < 512; k0 += 4) {
    v2f bf = *(const v2f*)(Wrow + k0 + kk);
    #pragma unroll
    for (int i = 0; i < 4; ++i) {
      v2f af = *(const v2f*)(Abase + (size_t)(m0 + i * 16 + mh) * 512 + k0 + kk);
      acc[i] = __builtin_amdgcn_wmma_f32_16x16x4_f32(
          false, af, false, bf, (short)0, acc[i], false, false);
    }
  }
}

// ---------------------------------------------------------------------------
// Generic GEMM: C[M,N] = A[M,512] @ W[N,512]^T + bias[N]   (EPI=0)
//               C = sigmoid(acc+bias) * aux                (EPI=1, gated ctx)
// grid = (N/64, M/128), block = 256 (8 waves).
// wave w: rows m0 = by*128 + (w&1)*64 (4 tiles), cols n0 = bx*64 + (w>>1)*16.
// ---------------------------------------------------------------------------
template <int EPI>
__global__ __launch_bounds__(256) void gemm_wmma_k512(
    const float* __restrict__ Amat, const float* __restrict__ Wmat,
    const float* __restrict__ bias, const float* __restrict__ aux,
    float* __restrict__ Cmat, int N) {
  const int tid = threadIdx.x, lane = tid & 31, wv = tid >> 5;
  const int mh = lane & 15, kk = (lane >> 4) << 1;
  const int n0 = blockIdx.x * 64 + ((wv >> 1) << 4);
  const int m0 = blockIdx.y * 128 + ((wv & 1) << 6);

  const float* wrow = Wmat + (size_t)(n0 + mh) * 512;
  __builtin_prefetch(wrow, 0, 3);  // global_prefetch_b8

  v8f acc[4] = {};
  wmma_k512(acc, Amat, wrow, m0, mh, kk);

  const int n = n0 + mh;
  const float bv = bias[n];
  const int mb = m0 + ((lane >> 4) << 3);
  #pragma unroll
  for (int i = 0; i < 4; ++i) {
    #pragma unroll
    for (int r = 0; r < 8; ++r) {
      size_t off = (size_t)(mb + i * 16 + r) * N + n;
      float val = acc[i][r] + bv;
      if (EPI == 1) val = sigmoidf_(val) * aux[off];
      Cmat[off] = val;
    }
  }
}

// ---------------------------------------------------------------------------
// LSTM gates GEMM: gates[128,2048] = [xemb|ctxg] @ W_lih^T + hn @ W_lhh^T + b
// K = 1536 split into three 512 segments. grid = (2048/64, 1), block = 256.
// ---------------------------------------------------------------------------
__global__ __launch_bounds__(256) void lstm_gates_kernel(
    const float* __restrict__ xemb, const float* __restrict__ ctxg,
    const float* __restrict__ hn, const float* __restrict__ W_lih,
    const float* __restrict__ W_lhh, const float* __restrict__ b_lih,
    const float* __restrict__ b_lhh, float* __restrict__ gates) {
  const int tid = threadIdx.x, lane = tid & 31, wv = tid >> 5;
  const int mh = lane & 15, kk = (lane >> 4) << 1;
  const int n0 = blockIdx.x * 64 + ((wv >> 1) << 4);
  const int m0 = (wv & 1) << 6;

  const float* wlih_row = W_lih + (size_t)(n0 + mh) * 1024;
  const float* wlhh_row = W_lhh + (size_t)(n0 + mh) * 512;
  __builtin_prefetch(wlih_row, 0, 3);

  v8f acc[4] = {};
  wmma_k512(acc, xemb, wlih_row, m0, mh, kk);         // k in [0,512)
  wmma_k512(acc, ctxg, wlih_row + 512, m0, mh, kk);   // k in [512,1024)
  wmma_k512(acc, hn, wlhh_row, m0, mh, kk);           // recurrent term

  const int n = n0 + mh;
  const float bv = b_lih[n] + b_lhh[n];
  const int mb = m0 + ((lane >> 4) << 3);
  #pragma unroll
  for (int i = 0; i < 4; ++i)
    #pragma unroll
    for (int r = 0; r < 8; ++r)
      gates[(size_t)(mb + i * 16 + r) * NGATE + n] = acc[i][r] + bv;
}

// ---------------------------------------------------------------------------
// Logits GEMM: logits[128,32000] = hn @ W_out^T + b_out
// Stores to out[(b*V + v)*T + t] and emits per-(row, 64-col-block) argmax
// partials. grid = (500,1), block = 256; block covers 128 rows x 64 cols.
// ---------------------------------------------------------------------------
__global__ __launch_bounds__(256) void logits_kernel(
    const float* __restrict__ hn, const float* __restrict__ W_out,
    const float* __restrict__ b_out, float* __restrict__ out,
    float* __restrict__ part_val, int* __restrict__ part_idx, int t) {
  __shared__ float tile[128][65];
  const int tid = threadIdx.x, lane = tid & 31, wv = tid >> 5;
  const int mh = lane & 15, kk = (lane >> 4) << 1;
  const int nbase = blockIdx.x * 64;
  const int lcolbase = (wv >> 1) << 4;
  const int n0 = nbase + lcolbase;
  const int m0 = (wv & 1) << 6;

  const float* wrow = W_out + (size_t)(n0 + mh) * 512;
  __builtin_prefetch(wrow, 0, 3);

  v8f acc[4] = {};
  wmma_k512(acc, hn, wrow, m0, mh, kk);

  const int lcol = lcolbase + mh;
  const float bv = b_out[nbase + lcol];
  const int mb = m0 + ((lane >> 4) << 3);
  #pragma unroll
  for (int i = 0; i < 4; ++i)
    #pragma unroll
    for (int r = 0; r < 8; ++r)
      tile[mb + i * 16 + r][lcol] = acc[i][r] + bv;
  __syncthreads();

  // Strided scatter into (B, V, T) output.
  for (int e = tid; e < 128 * 64; e += 256) {
    int row = e >> 6, col = e & 63;
    out[((size_t)row * VOC + nbase + col) * TSTEP + t] = tile[row][col];
  }
  // Per-row argmax over this 64-col block (strict > keeps first occurrence).
  if (tid < 128) {
    float best = tile[tid][0];
    int bi = nbase;
    #pragma unroll 4
    for (int c = 1; c < 64; ++c) {
      float v = tile[tid][c];
      if (v > best) { best = v; bi = nbase + c; }
    }
    part_val[tid * NPART + blockIdx.x] = best;
    part_idx[tid * NPART + blockIdx.x] = bi;
  }
}

// ---------------------------------------------------------------------------
// Reduce argmax partials for each batch row; gather next embedding.
// ---------------------------------------------------------------------------
__global__ __launch_bounds__(256) void argmax_embed_kernel(
    const float* __restrict__ part_val, const int* __restrict__ part_idx,
    const float* __restrict__ emb_table, float* __restrict__ xemb) {
  __shared__ float sv[256];
  __shared__ int si[256];
  const int b = blockIdx.x, tid = threadIdx.x;
  float best = -3.4e38f;
  int bi = 0x7fffffff;
  for (int i = tid; i < NPART; i += 256) {
    float v = part_val[b * NPART + i];
    int id = part_idx[b * NPART + i];
    if (v > best || (v == best && id < bi)) { best = v; bi = id; }
  }
  sv[tid] = best; si[tid] = bi;
  __syncthreads();
  for (int s = 128; s > 0; s >>= 1) {
    if (tid < s) {
      if (sv[tid + s] > sv[tid] ||
          (sv[tid + s] == sv[tid] && si[tid + s] < si[tid])) {
        sv[tid] = sv[tid + s]; si[tid] = si[tid + s];
      }
    }
    __syncthreads();
  }
  const int pred = si[0];
  for (int a = tid; a < EMBD; a += 256)
    xemb[(size_t)b * EMBD + a] = emb_table[(size_t)pred * EMBD + a];
}

// ---------------------------------------------------------------------------
// Attention: q precomputed. e = relu(ka + q) @ W_att^T + b_att; softmax over
// PIX; ctx = sum_p alpha * feature. One block per batch element.
// ---------------------------------------------------------------------------
__global__ __launch_bounds__(256) void attn_kernel(
    const float* __restrict__ ka, const float* __restrict__ q,
    const float* __restrict__ feature, const float* __restrict__ W_att,
    const float* __restrict__ b_att, float* __restrict__ ctx) {
  __shared__ float sq[ADIM], sw[ADIM], se[PIX], red[256];
  const int b = blockIdx.x, tid = threadIdx.x;
  const int lane = tid & 31, wv = tid >> 5;
  const float batt = b_att[0];

  for (int a = tid; a < ADIM; a += 256) {
    sq[a] = q[(size_t)b * ADIM + a];
    sw[a] = W_att[a];
  }
  __syncthreads();

  // scores
  for (int p = wv; p < PIX; p += 8) {
    const float* row = ka + ((size_t)b * PIX + p) * ADIM;
    float s = 0.f;
    for (int a = lane; a < ADIM; a += 32)
      s += fmaxf(row[a] + sq[a], 0.f) * sw[a];
    #pragma unroll
    for (int off = 16; off > 0; off >>= 1) s += __shfl_xor(s, off);
    if (lane == 0) se[p] = s + batt;
  }
  __syncthreads();

  // softmax max
  float mx = -3.4e38f;
  for (int p = tid; p < PIX; p += 256) mx = fmaxf(mx, se[p]);
  red[tid] = mx;
  __syncthreads();
  for (int s = 128; s > 0; s >>= 1) {
    if (tid < s) red[tid] = fmaxf(red[tid], red[tid + s]);
    __syncthreads();
  }
  mx = red[0];
  __syncthreads();
  // exp + sum
  float sm = 0.f;
  for (int p = tid; p < PIX; p += 256) {
    float ex = __expf(se[p] - mx);
    se[p] = ex;
    sm += ex;
  }
  red[tid] = sm;
  __syncthreads();
  for (int s = 128; s > 0; s >>= 1) {
    if (tid < s) red[tid] += red[tid + s];
    __syncthreads();
  }
  const float inv = 1.0f / red[0];

  // context
  for (int a = tid; a < EDIM; a += 256) {
    float acc = 0.f;
    const float* fb = feature + (size_t)b * PIX * EDIM + a;
    for (int p = 0; p < PIX; ++p) acc += se[p] * fb[(size_t)p * EDIM];
    ctx[(size_t)b * EDIM + a] = acc * inv;
  }
}

__global__ __launch_bounds__(256) void mean_kernel(
    const float* __restrict__ feat, float* __restrict__ mean_ft) {
  const int b = blockIdx.x, tid = threadIdx.x;
  for (int a = tid; a < EDIM; a += 256) {
    float s = 0.f;
    const float* fb = feat + (size_t)b * PIX * EDIM + a;
    for (int p = 0; p < PIX; ++p) s += fb[(size_t)p * EDIM];
    mean_ft[(size_t)b * EDIM + a] = s * (1.0f / PIX);
  }
}

__global__ __launch_bounds__(256) void embed0_kernel(
    const int* __restrict__ captions, const float* __restrict__ emb_table,
    float* __restrict__ xemb) {
  const int b = blockIdx.x, tid = threadIdx.x;
  const int tok = captions[b * TSTEP];
  for (int a = tid; a < EMBD; a += 256)
    xemb[(size_t)b * EMBD + a] = emb_table[(size_t)tok * EMBD + a];
}

__global__ __launch_bounds__(256) void lstm_update_kernel(
    const float* __restrict__ gates, float* __restrict__ hn,
    float* __restrict__ cn) {
  const int idx = blockIdx.x * 256 + threadIdx.x;  // 0 .. 128*512-1
  const int b = idx >> 9, n = idx & 511;
  const float* g = gates + (size_t)b * NGATE;
  float ig = sigmoidf_(g[n]);
  float fg = sigmoidf_(g[512 + n]);
  float gc = tanhf(g[1024 + n]);
  float og = sigmoidf_(g[1536 + n]);
  float c = fg * cn[idx] + ig * gc;
  cn[idx] = c;
  hn[idx] = og * tanhf(c);
}

// ---------------------------------------------------------------------------
extern "C" void kernel_launch(void* const* d_in, const int* in_sizes, int n_in,
                              void* d_out, int out_size, void* d_ws,
                              size_t ws_size, hipStream_t stream) {
  (void)in_sizes; (void)n_in; (void)out_size; (void)ws_size;
  const float* image_feature = (const float*)d_in[0];
  const int*   captions      = (const int*)d_in[1];
  const float* W_ienc  = (const float*)d_in[2];
  const float* b_ienc  = (const float*)d_in[3];
  const float* W_oenc  = (const float*)d_in[4];
  const float* b_oenc  = (const float*)d_in[5];
  const float* W_att   = (const float*)d_in[6];
  const float* b_att   = (const float*)d_in[7];
  const float* W_inith = (const float*)d_in[8];
  const float* b_inith = (const float*)d_in[9];
  const float* W_initc = (const float*)d_in[10];
  const float* b_initc = (const float*)d_in[11];
  const float* W_gate  = (const float*)d_in[12];
  const float* b_gate  = (const float*)d_in[13];
  const float* emb_table = (const float*)d_in[14];
  const float* W_lih   = (const float*)d_in[15];
  const float* W_lhh   = (const float*)d_in[16];
  const float* b_lih   = (const float*)d_in[17];
  const float* b_lhh   = (const float*)d_in[18];
  const float* W_out   = (const float*)d_in[19];
  const float* b_out   = (const float*)d_in[20];
  float* out = (float*)d_out;

  // Workspace carve-up (floats)
  float* ws = (float*)d_ws;
  size_t o = 0;
  float* mean_ft = ws + o; o += (size_t)BSZ * EDIM;
  float* hn      = ws + o; o += (size_t)BSZ * HDIM;
  float* cn      = ws + o; o += (size_t)BSZ * HDIM;
  float* xemb    = ws + o; o += (size_t)BSZ * EMBD;
  float* q       = ws + o; o += (size_t)BSZ * ADIM;
  float* ctx     = ws + o; o += (size_t)BSZ * EDIM;
  float* ctxg    = ws + o; o += (size_t)BSZ * EDIM;
  float* gates   = ws + o; o += (size_t)BSZ * NGATE;
  float* ka      = ws + o; o += (size_t)BSZ * PIX * ADIM;
  float* part_val = ws + o; o += (size_t)BSZ * NPART;
  int*   part_idx = (int*)(ws + o);

  // --- Prologue ---
  mean_kernel<<<BSZ, 256, 0, stream>>>(image_feature, mean_ft);
  gemm_wmma_k512<0><<<dim3(HDIM / 64, 1), 256, 0, stream>>>(
      mean_ft, W_inith, b_inith, nullptr, hn, HDIM);
  gemm_wmma_k512<0><<<dim3(HDIM / 64, 1), 256, 0, stream>>>(
      mean_ft, W_initc, b_initc, nullptr, cn, HDIM);
  gemm_wmma_k512<0><<<dim3(ADIM / 64, (BSZ * PIX) / 128), 256, 0, stream>>>(
      image_feature, W_ienc, b_ienc, nullptr, ka, ADIM);
  embed0_kernel<<<BSZ, 256, 0, stream>>>(captions, emb_table, xemb);

  // --- Recurrent steps ---
  for (int t = 0; t < TSTEP; ++t) {
    gemm_wmma_k512<0><<<dim3(ADIM / 64, 1), 256, 0, stream>>>(
        hn, W_oenc, b_oenc, nullptr, q, ADIM);
    attn_kernel<<<BSZ, 256, 0, stream>>>(ka, q, image_feature, W_att, b_att,
                                         ctx);
    gemm_wmma_k512<1><<<dim3(HDIM / 64, 1), 256, 0, stream>>>(
        ctx, W_gate, b_gate, ctx, ctxg, HDIM);
    lstm_gates_kernel<<<dim3(NGATE / 64, 1), 256, 0, stream>>>(
        xemb, ctxg, hn, W_lih, W_lhh, b_lih, b_lhh, gates);
    lstm_update_kernel<<<(BSZ * HDIM) / 256, 256, 0, stream>>>(gates, hn, cn);
    logits_kernel<<<dim3(NPART, 1), 256, 0, stream>>>(hn, W_out, b_out, out,
                                                      part_val, part_idx, t);
    argmax_embed_kernel<<<BSZ, 256, 0, stream>>>(part_val, part_idx, emb_table,
                                                 xemb);
  }
}